// AttnCoef_17351667876288
// MI455X (gfx1250) — compile-verified
//
#include <hip/hip_runtime.h>
#include <hip/hip_bf16.h>

// ---- problem constants (from reference: B=4, n=128, DIM_QK=64, N_HEADS=4) ----
#define NH     4        // heads
#define NB     4        // batch
#define NQ     128      // Lq = n
#define LK     16384    // Lk = n*n
#define DH     16       // head dim
#define NWAVES 16       // 512 threads / 32
#define NTILES (LK / 16) // 1024 column tiles of width 16

#define NEGBIG (-1.0e30f)   // finite "masked" sentinel: no NaN paths, exp underflows to 0

typedef float v2f __attribute__((ext_vector_type(2)));
typedef float v8f __attribute__((ext_vector_type(8)));

// merge two (max, sum-of-exp) softmax partials -- branchless with finite sentinel
__device__ __forceinline__ void combine_ms(float& m, float& s, float om, float os) {
    float nm = fmaxf(m, om);
    s = s * __expf(m - nm) + os * __expf(om - nm);
    m = nm;
}

// One 16x16 f32 sdp tile = Q_tile (16xDH) * K_tile^T (DHx16) via
// 4 chained V_WMMA_F32_16X16X4_F32 (K = 16 split into 4 chunks of 4).
__device__ __forceinline__ v8f qk_tile(const v2f aQ[4], const float* __restrict__ krow, int hi) {
    // B-matrix 4x16 f32 layout (mirrors documented A 16x4 layout):
    //  VGPR0: K=2*hi, VGPR1: K=2*hi+1 within each 4-chunk; N = lane%16.
    v2f bK[4];
#pragma unroll
    for (int t = 0; t < 4; ++t) {
        int d0 = 4 * t + 2 * hi;
        bK[t].x = krow[d0];
        bK[t].y = krow[d0 + 1];
    }
    v8f acc = {0.f, 0.f, 0.f, 0.f, 0.f, 0.f, 0.f, 0.f};
#pragma unroll
    for (int t = 0; t < 4; ++t) {
        acc = __builtin_amdgcn_wmma_f32_16x16x4_f32(
            /*neg_a=*/false, aQ[t], /*neg_b=*/false, bK[t],
            /*c_mod=*/(short)0, acc, /*reuse_a=*/false, /*reuse_b=*/false);
    }
    return acc;
}

__global__ __launch_bounds__(32 * NWAVES)
void attn_coef_kernel(const float* __restrict__ qA,   // [B, NQ, NH*DH]
                      const float* __restrict__ kA,   // [B, LK, NH*DH]
                      const int*   __restrict__ qmask,// [B, NQ]   (bool)
                      const int*   __restrict__ kmask,// [B, LK]   (bool, = node[j]&node[kk])
                      float* __restrict__ out)        // [NH, B, NQ, LK]
{
    const int qt   = blockIdx.x;          // 0..7  : 16-row q tile
    const int b    = blockIdx.y;          // 0..3
    const int h    = blockIdx.z;          // 0..3
    const int tid  = threadIdx.x;
    const int wave = tid >> 5;
    const int lane = tid & 31;
    const int col  = lane & 15;           // N position within tile
    const int hi   = lane >> 4;           // half-wave select
    const int i_base = qt * 16;

    __shared__ float lds_m[NWAVES][16];
    __shared__ float lds_s[NWAVES][16];
    __shared__ float lds_M[16];
    __shared__ float lds_RS[16];          // reciprocal of row sum (0 if fully masked)

    // ---- load A (Q tile, 16xDH f32) into WMMA 16x16x4 A layout ----
    // VGPR0 = K = 4t+2*hi, VGPR1 = K = 4t+2*hi+1 ; M = lane%16
    v2f aQ[4];
    {
        const int i = i_base + col;
        const float* qrow = qA + ((size_t)b * NQ + i) * (NH * DH) + h * DH;
#pragma unroll
        for (int t = 0; t < 4; ++t) {
            int d0 = 4 * t + 2 * hi;
            aQ[t].x = qrow[d0];
            aQ[t].y = qrow[d0 + 1];
        }
    }

    // C/D layout: VGPR v holds row M = v + 8*hi for this lane, column = col.
    bool qv[8];
#pragma unroll
    for (int v = 0; v < 8; ++v)
        qv[v] = qmask[b * NQ + i_base + v + 8 * hi] != 0;

    // ================= pass 1: masked online softmax stats (branchless) =================
    float m_run[8], s_run[8];
#pragma unroll
    for (int v = 0; v < 8; ++v) { m_run[v] = NEGBIG; s_run[v] = 0.0f; }

#pragma unroll 2
    for (int tile = wave; tile < NTILES; tile += NWAVES) {
        const int lk0 = tile * 16;
        const int lk  = lk0 + col;
        const int j   = lk0 >> 7;            // wave-uniform within the 16-wide tile
        const int kk  = (lk0 & 127) + col;
        const float* krow = kA + ((size_t)b * LK + lk) * (NH * DH) + h * DH;

        v8f acc = qk_tile(aQ, krow, hi);

        const bool colv = (kmask[(size_t)b * LK + lk] != 0) & (j != kk);
#pragma unroll
        for (int v = 0; v < 8; ++v) {
            const int i = i_base + v + 8 * hi;
            const bool valid = colv & qv[v] & (i != j) & (i != kk);
            float val = valid ? acc[v] * 0.25f : NEGBIG;     // 1/sqrt(DH) = 0.25
            float nm  = fmaxf(m_run[v], val);
            float p   = valid ? __expf(val - nm) : 0.0f;     // avoid exp(0)=1 on masked rows
            s_run[v]  = s_run[v] * __expf(m_run[v] - nm) + p;
            m_run[v]  = nm;
        }
    }

    // reduce across the 16 lanes of each half-wave (columns of the strip)
#pragma unroll
    for (int v = 0; v < 8; ++v) {
        float m = m_run[v], s = s_run[v];
#pragma unroll
        for (int off = 1; off < 16; off <<= 1) {
            float om = __shfl_xor(m, off, 32);
            float os = __shfl_xor(s, off, 32);
            combine_ms(m, s, om, os);
        }
        m_run[v] = m; s_run[v] = s;
    }
    if (col == 0) {
#pragma unroll
        for (int v = 0; v < 8; ++v) {
            lds_m[wave][v + 8 * hi] = m_run[v];
            lds_s[wave][v + 8 * hi] = s_run[v];
        }
    }
    __syncthreads();

    // merge per-wave partials -> final per-row (M, 1/S)
    if (tid < 16) {
        float m = NEGBIG, s = 0.0f;
#pragma unroll
        for (int w = 0; w < NWAVES; ++w)
            combine_ms(m, s, lds_m[w][tid], lds_s[w][tid]);
        lds_M[tid]  = m;
        lds_RS[tid] = (s > 0.0f) ? (1.0f / s) : 0.0f;   // fully-masked row -> 0 (matches ref)
    }
    __syncthreads();

    float rowM[8], rowRS[8];
#pragma unroll
    for (int v = 0; v < 8; ++v) {
        rowM[v]  = lds_M[v + 8 * hi];
        rowRS[v] = lds_RS[v + 8 * hi];
    }

    // ================= pass 2: recompute tiles, write alpha once =================
    float* outhb = out + (((size_t)h * NB + b) * NQ) * LK;
#pragma unroll 2
    for (int tile = wave; tile < NTILES; tile += NWAVES) {
        const int lk0 = tile * 16;
        const int lk  = lk0 + col;
        const int j   = lk0 >> 7;
        const int kk  = (lk0 & 127) + col;
        const float* krow = kA + ((size_t)b * LK + lk) * (NH * DH) + h * DH;

        v8f acc = qk_tile(aQ, krow, hi);

        const bool colv = (kmask[(size_t)b * LK + lk] != 0) & (j != kk);
#pragma unroll
        for (int v = 0; v < 8; ++v) {
            const int i = i_base + v + 8 * hi;
            const bool valid = colv & qv[v] & (i != j) & (i != kk);
            float e = __expf(acc[v] * 0.25f - rowM[v]) * rowRS[v];
            outhb[(size_t)i * LK + lk] = valid ? e : 0.0f;
        }
    }
}

extern "C" void kernel_launch(void* const* d_in, const int* in_sizes, int n_in,
                              void* d_out, int out_size, void* d_ws, size_t ws_size,
                              hipStream_t stream) {
    const float* qA    = (const float*)d_in[0]; // [4,128,64]
    const float* kA    = (const float*)d_in[1]; // [4,16384,64]
    const int*   qmask = (const int*)d_in[2];   // [4,128] bool
    const int*   kmask = (const int*)d_in[3];   // [4,128,128] bool
    float* out = (float*)d_out;                 // [4,4,128,16384]

    dim3 grid(NQ / 16, NB, NH);                 // (8, 4, 4)
    dim3 block(32 * NWAVES);                    // 512 threads = 16 waves
    attn_coef_kernel<<<grid, block, 0, stream>>>(qA, kA, qmask, kmask, out);
}